// VanillaGCN_79645873537297
// MI455X (gfx1250) — compile-verified
//
#include <hip/hip_runtime.h>
#include <math.h>

typedef __attribute__((ext_vector_type(2))) float v2f;
typedef __attribute__((ext_vector_type(8))) float v8f;

#define N_NODES 50000
#define N_EDGES 800000
#define IN_DIM  128
#define HID     64
#define NCLASS  32

// ---------------------------------------------------------------------------
// small utility kernels
// ---------------------------------------------------------------------------
__global__ void fill_f32(float* __restrict__ p, float v, int n) {
    int i = blockIdx.x * blockDim.x + threadIdx.x;
    if (i < n) p[i] = v;
}

__global__ void edge_degree(const long long* __restrict__ ei, float* __restrict__ deg) {
    int e = blockIdx.x * blockDim.x + threadIdx.x;
    if (e < N_EDGES) {
        long long d = ei[N_EDGES + e];
        atomicAdd(&deg[d], 1.0f);
    }
}

__global__ void make_dinv(float* __restrict__ deg) {
    int i = blockIdx.x * blockDim.x + threadIdx.x;
    if (i < N_NODES) deg[i] = rsqrtf(deg[i]);
}

// one wave per edge: lane handles 2 contiguous features (float2 = 64 feats/wave)
__global__ void edge_scatter(const long long* __restrict__ ei,
                             const float* __restrict__ xw,
                             const float* __restrict__ dinv,
                             float* __restrict__ agg) {
    int gtid = blockIdx.x * blockDim.x + threadIdx.x;
    int wave = gtid >> 5;
    int lane = threadIdx.x & 31;
    if (wave >= N_EDGES) return;
    long long s = ei[wave];
    long long d = ei[N_EDGES + wave];
    float coef = dinv[s] * dinv[d];
    const float2* row = (const float2*)(xw + (long long)s * HID);
    float2 v = row[lane];
    float* dst = agg + (long long)d * HID + lane * 2;
    atomicAdd(dst + 0, v.x * coef);
    atomicAdd(dst + 1, v.y * coef);
}

// agg = (tanh?)(agg + xw * dinv^2 + bias[f])   (elementwise over N*HID)
__global__ void finalize_layer(const float* __restrict__ xw,
                               const float* __restrict__ bias,
                               const float* __restrict__ dinv,
                               float* __restrict__ agg,
                               int apply_tanh) {
    int i = blockIdx.x * blockDim.x + threadIdx.x;
    if (i >= N_NODES * HID) return;
    int node = i / HID;
    int f    = i % HID;
    float di = dinv[node];
    float v  = agg[i] + xw[i] * (di * di) + bias[f];
    agg[i] = apply_tanh ? tanhf(v) : v;
}

// per-node log_softmax over 32 classes
__global__ void log_softmax32(const float* __restrict__ logits, float* __restrict__ out) {
    int i = blockIdx.x * blockDim.x + threadIdx.x;
    if (i >= N_NODES) return;
    const float* r = logits + (long long)i * NCLASS;
    float mx = r[0];
    #pragma unroll
    for (int c = 1; c < NCLASS; ++c) mx = fmaxf(mx, r[c]);
    float s = 0.0f;
    #pragma unroll
    for (int c = 0; c < NCLASS; ++c) s += expf(r[c] - mx);
    float lse = logf(s) + mx;
    float* o = out + (long long)i * NCLASS;
    #pragma unroll
    for (int c = 0; c < NCLASS; ++c) o[c] = r[c] - lse;
}

// ---------------------------------------------------------------------------
// fp32 WMMA GEMM: out[M,NC] = op(A)[M,K] @ W[K,NC] (+ bias)
// op = tanh elementwise if TANH_A. One wave computes a 16-row x NC stripe.
// A frag (16x4 f32): lane m=lane&15 holds {A[m][kb], A[m][kb+1]}, kb=2*(lane>>4)
// B frag (4x16 f32): lane n=lane&15 holds {W[kb][n], W[kb+1][n]}
// C/D (16x16 f32, 8 VGPRs): vgpr i -> row i (lanes 0-15) / row i+8 (lanes 16-31)
// ---------------------------------------------------------------------------
template <int K, int NC, bool TANH_A, bool ADD_BIAS>
__global__ void wmma_gemm_f32(const float* __restrict__ A,
                              const float* __restrict__ W,
                              const float* __restrict__ bias,
                              float* __restrict__ out,
                              int M) {
    int gtid = blockIdx.x * blockDim.x + threadIdx.x;
    int wave = gtid >> 5;
    int lane = threadIdx.x & 31;
    int mtiles = M >> 4;
    if (wave >= mtiles) return;

    const int row0  = wave << 4;
    const int m     = lane & 15;
    const int khalf = (lane >> 4) << 1;      // 0 or 2
    constexpr int NT = NC / 16;

    v8f acc[NT];
    const v8f zero = {0.f, 0.f, 0.f, 0.f, 0.f, 0.f, 0.f, 0.f};
    #pragma unroll
    for (int j = 0; j < NT; ++j) acc[j] = zero;

    const float* arow = A + (long long)(row0 + m) * K;

    #pragma unroll 4
    for (int k = 0; k < K; k += 4) {
        float a0 = arow[k + khalf + 0];
        float a1 = arow[k + khalf + 1];
        if (TANH_A) { a0 = tanhf(a0); a1 = tanhf(a1); }
        v2f a; a.x = a0; a.y = a1;
        #pragma unroll
        for (int j = 0; j < NT; ++j) {
            int col = j * 16 + m;
            v2f b;
            b.x = W[(k + khalf + 0) * NC + col];
            b.y = W[(k + khalf + 1) * NC + col];
            acc[j] = __builtin_amdgcn_wmma_f32_16x16x4_f32(
                false, a, false, b, (short)0, acc[j], false, false);
        }
    }

    const int rbase = row0 + ((lane >> 4) << 3);   // +8 for upper half-wave
    #pragma unroll
    for (int j = 0; j < NT; ++j) {
        int col = j * 16 + m;
        float bv = ADD_BIAS ? bias[col] : 0.0f;
        #pragma unroll
        for (int i = 0; i < 8; ++i) {
            out[(long long)(rbase + i) * NC + col] = acc[j][i] + bv;
        }
    }
}

// ---------------------------------------------------------------------------
extern "C" void kernel_launch(void* const* d_in, const int* in_sizes, int n_in,
                              void* d_out, int out_size, void* d_ws, size_t ws_size,
                              hipStream_t stream) {
    (void)in_sizes; (void)n_in; (void)out_size; (void)ws_size;

    const float*     x   = (const float*)d_in[0];
    const long long* ei  = (const long long*)d_in[1];   // int64 edge_index [2, E]
    const float*     W1  = (const float*)d_in[2];
    const float*     b1  = (const float*)d_in[3];
    const float*     W2  = (const float*)d_in[4];
    const float*     b2  = (const float*)d_in[5];
    const float*     Wp1 = (const float*)d_in[6];
    const float*     bp1 = (const float*)d_in[7];
    const float*     Wp2 = (const float*)d_in[8];
    const float*     bp2 = (const float*)d_in[9];

    float* emb = (float*)d_out;                               // [N, 64]
    float* lsm = (float*)d_out + (size_t)N_NODES * HID;       // [N, 32]

    float* dinv = (float*)d_ws;                               // N
    float* bufA = dinv + N_NODES;                             // N*64
    float* bufB = bufA + (size_t)N_NODES * HID;               // N*64

    const int T = 256;
    const int gridN    = (N_NODES + T - 1) / T;
    const int gridE    = (N_EDGES + T - 1) / T;
    const int gridNH   = (N_NODES * HID + T - 1) / T;
    const int gridEdgeWaves = N_EDGES / (T / 32);             // 1 wave/edge
    const int mtiles   = N_NODES / 16;                        // 3125 exact
    const int gridGemm = (mtiles + (T / 32) - 1) / (T / 32);

    // degree -> dinv
    fill_f32<<<gridN, T, 0, stream>>>(dinv, 1.0f, N_NODES);   // self-loop +1
    edge_degree<<<gridE, T, 0, stream>>>(ei, dinv);
    make_dinv<<<gridN, T, 0, stream>>>(dinv);

    // layer 1: xw1 = x @ W1 ; agg ; h1 = tanh(agg + xw1*dinv^2 + b1)
    wmma_gemm_f32<IN_DIM, HID, false, false><<<gridGemm, T, 0, stream>>>(x, W1, nullptr, bufA, N_NODES);
    hipMemsetAsync(bufB, 0, (size_t)N_NODES * HID * sizeof(float), stream);
    edge_scatter<<<gridEdgeWaves, T, 0, stream>>>(ei, bufA, dinv, bufB);
    finalize_layer<<<gridNH, T, 0, stream>>>(bufA, b1, dinv, bufB, 1);

    // layer 2: xw2 = h1 @ W2 ; agg into d_out embedding ; emb = agg + xw2*dinv^2 + b2
    wmma_gemm_f32<HID, HID, false, false><<<gridGemm, T, 0, stream>>>(bufB, W2, nullptr, bufA, N_NODES);
    hipMemsetAsync(emb, 0, (size_t)N_NODES * HID * sizeof(float), stream);
    edge_scatter<<<gridEdgeWaves, T, 0, stream>>>(ei, bufA, dinv, emb);
    finalize_layer<<<gridNH, T, 0, stream>>>(bufA, b2, dinv, emb, 0);

    // head: h3 = tanh(emb) @ Wp1 + bp1 ; logits = h3 @ Wp2 + bp2 ; log_softmax
    wmma_gemm_f32<HID, HID, true, true><<<gridGemm, T, 0, stream>>>(emb, Wp1, bp1, bufB, N_NODES);
    wmma_gemm_f32<HID, NCLASS, false, true><<<gridGemm, T, 0, stream>>>(bufB, Wp2, bp2, bufA, N_NODES);
    log_softmax32<<<gridN, T, 0, stream>>>(bufA, lsm);
}